// OFTLinearLayer_20547123544604
// MI455X (gfx1250) — compile-verified
//
#include <hip/hip_runtime.h>
#include <hip/hip_bf16.h>
#include <stddef.h>
#include <stdint.h>

// OFT linear layer for MI455X (gfx1250), fp32 end-to-end.
//   out = (x @ blockdiag(Q)) @ W^T,  Q_r = (I+S_r)(I-S_r)^{-1},  S = 0.5(R - R^T)
// All GEMMs are NT (C = A * B^T, both operands row-major with contiguous K):
//   MinvT = ((I-S)^{-1})^T            (Gauss-Jordan writes transposed)
//   Qt    = NT(MinvT, P)   == Q^T
//   y     = NT(x, Qt)      == x @ blockdiag(Q)
//   out   = NT(y, W)       == y @ W^T
// Workspace (float): wsM(4MB) wsP(4MB) wsQt(4MB) wsY(64MB) -> 76 MiB total.
// GEMM: V_WMMA_F32_16X16X4_F32, double-buffered LDS tiles filled by
// GLOBAL_LOAD_ASYNC_TO_LDS_B128 (ASYNCcnt / s_wait_asynccnt).

typedef float v2f __attribute__((ext_vector_type(2)));
typedef float v8f __attribute__((ext_vector_type(8)));

#define RB   16
#define BLK  256
#define NTOK 4096
#define NF   4096
#define BLKSQ (BLK * BLK)

// ---------------------------------------------------------------------------
// gfx1250 async global->LDS copy (ASYNCcnt-tracked) + wait, via inline asm.
// ---------------------------------------------------------------------------
__device__ __forceinline__ void async_b128(const float* gsrc, const float* ldst) {
  const unsigned lds = (unsigned)(uintptr_t)ldst;
  asm volatile("global_load_async_to_lds_b128 %0, %1, off"
               :: "v"(lds), "v"(gsrc) : "memory");
}
__device__ __forceinline__ void async_wait0() {
  asm volatile("s_wait_asynccnt 0x0" ::: "memory");
}

// ---------------------------------------------------------------------------
// Kernel 1: M = I - S, P = I + S  (S = 0.5*(R - R^T) per 256x256 block)
// ---------------------------------------------------------------------------
__global__ __launch_bounds__(256) void oft_cayley_prep(
    const float* __restrict__ R, float* __restrict__ M, float* __restrict__ P) {
  const size_t idx = (size_t)blockIdx.x * 256u + threadIdx.x;
  const int r  = (int)(idx >> 16);
  const int ij = (int)(idx & 65535u);
  const int i  = ij >> 8;
  const int j  = ij & 255;
  const float* Rb = R + (size_t)r * BLKSQ;
  const float s = 0.5f * (Rb[i * BLK + j] - Rb[j * BLK + i]);
  const float d = (i == j) ? 1.0f : 0.0f;
  M[idx] = d - s;
  P[idx] = d + s;
}

// ---------------------------------------------------------------------------
// Kernel 2: in-place Gauss-Jordan inversion, one 256x256 block per WG in LDS;
// result written back TRANSPOSED. (I-S) has SPD symmetric part (=I) so no
// pivoting is required. Row stride 257 => column reads are bank-conflict-free.
// ---------------------------------------------------------------------------
#define GJ_STRIDE 257
__global__ __launch_bounds__(1024) void oft_gj_invert_t(float* __restrict__ Mats) {
  extern __shared__ float sA[];                 // 256*257 + 256 floats
  float* col = sA + BLK * GJ_STRIDE;
  float* A = Mats + (size_t)blockIdx.x * BLKSQ;

  for (int idx = threadIdx.x; idx < BLKSQ; idx += 1024)
    sA[(idx >> 8) * GJ_STRIDE + (idx & 255)] = A[idx];
  __syncthreads();

  const int j  = threadIdx.x & 255;
  const int rg = threadIdx.x >> 8;

  for (int k = 0; k < BLK; ++k) {
    if (rg == 0) col[j] = sA[j * GJ_STRIDE + k];
    __syncthreads();
    const float invd = 1.0f / col[k];
    if (rg == 0) {
      const float v = (j == k) ? invd : sA[k * GJ_STRIDE + j] * invd;
      sA[k * GJ_STRIDE + j] = v;
    }
    __syncthreads();
    const float rk = sA[k * GJ_STRIDE + j];
    for (int i = rg; i < BLK; i += 4) {
      if (i == k) continue;
      const float e   = col[i];
      const float cur = sA[i * GJ_STRIDE + j];
      sA[i * GJ_STRIDE + j] = (j == k) ? (-e * invd) : fmaf(-e, rk, cur);
    }
    __syncthreads();
  }

  // Transposed writeback: A[jo][io] = inv[io][jo]; LDS reads stride 257
  // (banks = io mod 64, conflict-free), global writes fully coalesced.
  for (int idx = threadIdx.x; idx < BLKSQ; idx += 1024) {
    const int jo = idx >> 8;
    const int io = idx & 255;
    A[idx] = sA[io * GJ_STRIDE + jo];
  }
}

// ---------------------------------------------------------------------------
// Kernel 3: batched fp32 NT WMMA GEMM, C[z] = A[z] * B[z]^T.
//   A row-major M x K (lda), B row-major N x K (ldb), C row-major M x N (ldc).
// WG = 256 threads = 8 waves laid out 4(M)x2(N); WG tile 128x128; each wave
// owns a 32x64 C tile (8 f32 accumulators). K consumed in chunks of 32,
// staged global->LDS with async b128 copies, double-buffered. LDS row stride
// 36 floats => 32-lane b64 fragment reads hit 64 distinct banks and the b128
// async fills stay 16B-aligned.
// ---------------------------------------------------------------------------
__global__ __launch_bounds__(256) void oft_wmma_gemm_nt(
    const float* __restrict__ A, int lda,
    const float* __restrict__ B, int ldb,
    float* __restrict__ C, int ldc, int K,
    int aStride, int bStride, int cStride) {
  constexpr int KC    = 32;
  constexpr int PAD   = 36;            // tile row stride (floats)
  constexpr int TSZ   = 128 * PAD;     // 4608 floats per tile
  constexpr int BUFSZ = 2 * TSZ;       // A tile + B tile
  extern __shared__ float smem[];      // 2 * BUFSZ floats

  const int tid  = threadIdx.x;
  const int wave = tid >> 5, lane = tid & 31;
  const int ln = lane & 15, lh = lane >> 4, kk = lh * 2;
  const int wm = wave >> 1, wn = wave & 1;

  const float* Ab = A + (size_t)blockIdx.z * aStride;
  const float* Bb = B + (size_t)blockIdx.z * bStride;
  float*       Cb = C + (size_t)blockIdx.z * cStride;
  const int row0 = blockIdx.y * 128;
  const int col0 = blockIdx.x * 128;

  const int frow = tid >> 3, fcol = (tid & 7) * 4;   // 128x32 tile fill coords

  auto prefetch = [&](int k0, int buf) {
    float* sA = smem + buf * BUFSZ;
    float* sB = sA + TSZ;
#pragma unroll
    for (int rr = 0; rr < 4; ++rr) {
      const int row = rr * 32 + frow;
      async_b128(Ab + (size_t)(row0 + row) * lda + k0 + fcol,
                 sA + row * PAD + fcol);
      async_b128(Bb + (size_t)(col0 + row) * ldb + k0 + fcol,
                 sB + row * PAD + fcol);
    }
  };

  v8f acc[2][4] = {};

  const int nChunks = K / KC;
  prefetch(0, 0);
  async_wait0();
  __syncthreads();

  for (int c = 0; c < nChunks; ++c) {
    if (c + 1 < nChunks) prefetch((c + 1) * KC, (c + 1) & 1);

    const float* sA = smem + (c & 1) * BUFSZ;
    const float* sB = sA + TSZ;
    const float* aF0 = sA + (wm * 32 + ln) * PAD + kk;
    const float* aF1 = aF0 + 16 * PAD;
    const float* bF  = sB + (wn * 64 + ln) * PAD + kk;

#pragma unroll
    for (int kc = 0; kc < KC; kc += 4) {
      const v2f a0 = *(const v2f*)(aF0 + kc);
      const v2f a1 = *(const v2f*)(aF1 + kc);
#pragma unroll
      for (int t = 0; t < 4; ++t) {
        const v2f b = *(const v2f*)(bF + t * (16 * PAD) + kc);
        acc[0][t] = __builtin_amdgcn_wmma_f32_16x16x4_f32(
            false, a0, false, b, (short)0, acc[0][t], false, false);
        acc[1][t] = __builtin_amdgcn_wmma_f32_16x16x4_f32(
            false, a1, false, b, (short)0, acc[1][t], false, false);
      }
    }
    async_wait0();     // next chunk's fills done (this wave's share)
    __syncthreads();   // all waves done with chunk c / fills of c+1
  }

  // C/D layout: VGPR g -> row (+g) lanes 0-15, row (+8+g) lanes 16-31
#pragma unroll
  for (int s = 0; s < 2; ++s) {
    const int rbase = row0 + wm * 32 + s * 16 + lh * 8;
#pragma unroll
    for (int t = 0; t < 4; ++t) {
      const int col = col0 + wn * 64 + t * 16 + ln;
#pragma unroll
      for (int g = 0; g < 8; ++g) {
        Cb[(size_t)(rbase + g) * ldc + col] = acc[s][t][g];
      }
    }
  }
}

// ---------------------------------------------------------------------------
extern "C" void kernel_launch(void* const* d_in, const int* in_sizes, int n_in,
                              void* d_out, int out_size, void* d_ws, size_t ws_size,
                              hipStream_t stream) {
  const float* x = (const float*)d_in[0];  // [4096, 4096]
  const float* W = (const float*)d_in[1];  // [4096, 4096] (out, in)
  const float* R = (const float*)d_in[2];  // [16, 256, 256]
  float* out = (float*)d_out;              // [4096, 4096]

  float* wsM  = (float*)d_ws;                // I - S  -> (I-S)^{-T}
  float* wsP  = wsM + (size_t)RB * BLKSQ;    // I + S
  float* wsQt = wsP + (size_t)RB * BLKSQ;    // Q^T
  float* wsY  = wsQt + (size_t)RB * BLKSQ;   // x @ blockdiag(Q)

  // 1) M = I - S, P = I + S
  oft_cayley_prep<<<(RB * BLKSQ) / 256, 256, 0, stream>>>(R, wsM, wsP);

  // 2) M <- (M^{-1})^T
  const size_t gjLds = (size_t)(BLK * GJ_STRIDE + BLK) * sizeof(float);
  oft_gj_invert_t<<<RB, 1024, gjLds, stream>>>(wsM);

  const size_t ldsNT = 2u * 2u * 128u * 36u * sizeof(float);  // 73728 B

  // 3) Qt = MinvT * P^T  (== Q^T) : 16 batched 256x256x256 NT GEMMs
  oft_wmma_gemm_nt<<<dim3(BLK / 128, BLK / 128, RB), 256, ldsNT, stream>>>(
      wsM, BLK, wsP, BLK, wsQt, BLK, BLK, BLKSQ, BLKSQ, BLKSQ);

  // 4) y = x * Qt^T (== x @ blockdiag(Q)) : 16 batched 4096x256x256 NT GEMMs
  oft_wmma_gemm_nt<<<dim3(BLK / 128, NTOK / 128, RB), 256, ldsNT, stream>>>(
      x, NF, wsQt, BLK, wsY, NF, BLK, BLK, BLKSQ, BLK);

  // 5) out = y * W^T : 4096x4096x4096 NT GEMM
  oft_wmma_gemm_nt<<<dim3(NF / 128, NTOK / 128, 1), 256, ldsNT, stream>>>(
      wsY, NF, W, NF, out, NF, NF, 0, 0, 0);
}